// Block2x2DiagProduct_50397146251620
// MI455X (gfx1250) — compile-verified
//
#include <hip/hip_runtime.h>

typedef float v2f __attribute__((ext_vector_type(2)));
typedef float v8f __attribute__((ext_vector_type(8)));

#define ROWS    16
#define TSTR    1028            // tile row stride in floats: %64==4 -> conflict-free, rows 16B aligned
#define MSTR    17              // matrix row stride (bank-conflict-free for lane=row reads)
#define MATSZ   (16 * MSTR)     // floats per 16x16 matrix slot
#define NMATS   17              // T0 + 16 * T1[b]
#define THREADS 256

// Apply one butterfly stage (half = h) to a 16-vector; coeff index = kk. (for T0: stages 9..6)
__device__ __forceinline__ void apply_stage_lo(float v[16], const float* __restrict__ p, const int h) {
#pragma unroll
    for (int B0 = 0; B0 < 16; B0 += 2 * h) {
#pragma unroll
        for (int kk = 0; kk < h; ++kk) {
            const float a = p[kk], b = p[h + kk], c = p[2 * h + kk], d = p[3 * h + kk];
            const float t = v[B0 + kk], u = v[B0 + kk + h];
            v[B0 + kk]     = a * t + b * u;
            v[B0 + kk + h] = c * t + d * u;
        }
    }
}

// Apply one stage in the stride-16 digit space; coeff index = kk*16 + b, stage half = h*16. (for T1: stages 5..2)
__device__ __forceinline__ void apply_stage_hi(float v[16], const float* __restrict__ p, const int h, const int b) {
    const int hf = h * 16;
#pragma unroll
    for (int B0 = 0; B0 < 16; B0 += 2 * h) {
#pragma unroll
        for (int kk = 0; kk < h; ++kk) {
            const int k = kk * 16 + b;
            const float a = p[k], bb = p[hf + k], c = p[2 * hf + k], d = p[3 * hf + k];
            const float t = v[B0 + kk], u = v[B0 + kk + h];
            v[B0 + kk]     = a * t + bb * u;
            v[B0 + kk + h] = c * t + d * u;
        }
    }
}

__global__ __launch_bounds__(THREADS)
void butterfly_wmma_kernel(const float* __restrict__ x,
                           const float* __restrict__ p0, const float* __restrict__ p1,
                           const float* __restrict__ p2, const float* __restrict__ p3,
                           const float* __restrict__ p4, const float* __restrict__ p5,
                           const float* __restrict__ p6, const float* __restrict__ p7,
                           const float* __restrict__ p8, const float* __restrict__ p9,
                           float* __restrict__ out, int nrows)
{
    __shared__ float tile[ROWS * TSTR];     // 65792 B
    __shared__ float mats[NMATS * MATSZ];   // 18496 B

    const int tid  = threadIdx.x;
    const int lane = tid & 31;
    const int wave = tid >> 5;
    const int n    = lane & 15;     // N index (batch row) == M index for A operand
    const int hi   = lane >> 4;
    const long row0 = (long)blockIdx.x * ROWS;
    if (row0 + ROWS > nrows) return;

    // ---------------- phase 1 (issue first): async DMA global -> LDS for the 16x1024 tile.
    // GLOBAL_LOAD_ASYNC_TO_LDS_B128 (ASYNCcnt): bypasses VGPRs, overlaps with phase 0 below.
#pragma unroll
    for (int it = 0; it < 16; ++it) {
        const int idx = it * THREADS + tid;     // 0..4095 float4 tasks
        const int r   = idx >> 8;
        const int c4  = idx & 255;
        const unsigned long long gaddr =
            (unsigned long long)(const void*)(x + (row0 + r) * 1024 + c4 * 4);
        const unsigned int laddr =                      // LDS byte offset = low 32 bits of generic ptr
            (unsigned int)(unsigned long long)(const void*)(&tile[r * TSTR + c4 * 4]);
        asm volatile("global_load_async_to_lds_b128 %0, %1, off"
                     :: "v"(laddr), "v"(gaddr) : "memory");
    }

    // ---------------- phase 0: build T0 and T1[b] (columns) into LDS (overlaps the DMA) ----------------
    for (int task = tid; task < 16 + 256; task += THREADS) {
        float v[16];
        if (task < 16) {
            const int q = task;
#pragma unroll
            for (int e = 0; e < 16; ++e) v[e] = (e == q) ? 1.0f : 0.0f;
            apply_stage_lo(v, p9, 1);
            apply_stage_lo(v, p8, 2);
            apply_stage_lo(v, p7, 4);
            apply_stage_lo(v, p6, 8);
#pragma unroll
            for (int r = 0; r < 16; ++r) mats[r * MSTR + q] = v[r];
        } else {
            const int idx = task - 16;
            const int b = idx >> 4;     // low-4-bit digit
            const int q = idx & 15;     // column in d1-space
#pragma unroll
            for (int e = 0; e < 16; ++e) v[e] = (e == q) ? 1.0f : 0.0f;
            apply_stage_hi(v, p5, 1, b);
            apply_stage_hi(v, p4, 2, b);
            apply_stage_hi(v, p3, 4, b);
            apply_stage_hi(v, p2, 8, b);
#pragma unroll
            for (int r = 0; r < 16; ++r) mats[(1 + b) * MATSZ + r * MSTR + q] = v[r];
        }
    }

    asm volatile("s_wait_asynccnt 0x0" ::: "memory");   // tile DMA complete (this wave)
    __syncthreads();                                    // all waves' DMA + mats visible

    // ---------------- phase A: stages 9..6 == per-16-chunk GEMM with T0 ----------------
    // D[r_out, n] = sum_q T0[r_out, q] * X[n, c*16+q]; 64 chunks, 8 per wave.
    for (int cc = 0; cc < 8; ++cc) {
        const int c = wave * 8 + cc;
        const float* Arow = &mats[n * MSTR];                  // T0 row n (lane m == n)
        const float* Brow = &tile[n * TSTR + c * 16];
        v8f acc = {};
#pragma unroll
        for (int t = 0; t < 4; ++t) {
            const int kk = 4 * t + 2 * hi;
            v2f a = { Arow[kk], Arow[kk + 1] };
            v2f b = *(const v2f*)(Brow + kk);                 // contiguous 8B LDS load
            acc = __builtin_amdgcn_wmma_f32_16x16x4_f32(false, a, false, b,
                                                        (short)0, acc, false, false);
        }
        float* dst = &tile[n * TSTR + c * 16 + 8 * hi];       // D[j+8*hi, n] -> col c*16+8*hi+j
        float4 lo  = { acc[0], acc[1], acc[2], acc[3] };
        float4 hi4 = { acc[4], acc[5], acc[6], acc[7] };
        *(float4*)dst = lo;
        *(float4*)(dst + 4) = hi4;
    }
    __syncthreads();

    // ---------------- phase B: stages 5..2 == stride-16 GEMM with T1[b] ----------------
    // task (b, d2): D[d1o, n] = sum_d1 T1[b][d1o, d1] * Y[n, d2*256 + d1*16 + b]
    for (int cc = 0; cc < 8; ++cc) {
        const int tsk = wave * 8 + cc;
        const int b   = tsk & 15;
        const int d2  = tsk >> 4;
        const float* Arow  = &mats[(1 + b) * MATSZ + n * MSTR];
        const float* Bbase = &tile[n * TSTR + d2 * 256 + b];
        v8f acc = {};
#pragma unroll
        for (int t = 0; t < 4; ++t) {
            const int kk = 4 * t + 2 * hi;
            v2f a  = { Arow[kk], Arow[kk + 1] };
            v2f bb = { Bbase[kk * 16], Bbase[kk * 16 + 16] };
            acc = __builtin_amdgcn_wmma_f32_16x16x4_f32(false, a, false, bb,
                                                        (short)0, acc, false, false);
        }
        float* dst = &tile[n * TSTR + d2 * 256 + 8 * hi * 16 + b];
#pragma unroll
        for (int j = 0; j < 8; ++j) dst[j * 16] = acc[j];
    }
    __syncthreads();

    // ---------------- phase C: stages 1 and 0 on VALU, fused with coalesced store ----------------
    const int k = tid;                                        // 0..255
    const float a1 = p1[k],        b1 = p1[256 + k],   c1 = p1[512 + k],   d1c = p1[768 + k];
    const float a0 = p0[k],        b0 = p0[512 + k],   c0 = p0[1024 + k],  d0  = p0[1536 + k];
    const float a0b = p0[256 + k], b0b = p0[768 + k],  c0b = p0[1280 + k], d0b = p0[1792 + k];
#pragma unroll
    for (int r = 0; r < 16; ++r) {
        const float z0 = tile[r * TSTR + k];
        const float z1 = tile[r * TSTR + k + 256];
        const float z2 = tile[r * TSTR + k + 512];
        const float z3 = tile[r * TSTR + k + 768];
        // stage i=1 (s=512): pairs (k, k+256) and (512+k, 768+k), coeff index k
        const float y0 = a1 * z0 + b1 * z1;
        const float y1 = c1 * z0 + d1c * z1;
        const float y2 = a1 * z2 + b1 * z3;
        const float y3 = c1 * z2 + d1c * z3;
        // stage i=0 (s=1024): pairs (k, k+512) coeff k ; (k+256, k+768) coeff k+256
        const float w0 = a0 * y0 + b0 * y2;
        const float w2 = c0 * y0 + d0 * y2;
        const float w1 = a0b * y1 + b0b * y3;
        const float w3 = c0b * y1 + d0b * y3;
        float* o = out + (row0 + r) * 1024;
        o[k]       = w0;
        o[k + 256] = w1;
        o[k + 512] = w2;
        o[k + 768] = w3;
    }
}

extern "C" void kernel_launch(void* const* d_in, const int* in_sizes, int n_in,
                              void* d_out, int out_size, void* d_ws, size_t ws_size,
                              hipStream_t stream) {
    (void)n_in; (void)out_size; (void)d_ws; (void)ws_size;
    const float* x  = (const float*)d_in[0];
    const float* p0 = (const float*)d_in[1];
    const float* p1 = (const float*)d_in[2];
    const float* p2 = (const float*)d_in[3];
    const float* p3 = (const float*)d_in[4];
    const float* p4 = (const float*)d_in[5];
    const float* p5 = (const float*)d_in[6];
    const float* p6 = (const float*)d_in[7];
    const float* p7 = (const float*)d_in[8];
    const float* p8 = (const float*)d_in[9];
    const float* p9 = (const float*)d_in[10];
    float* out = (float*)d_out;
    const int nrows = in_sizes[0] / 1024;
    const int grid  = nrows / ROWS;
    butterfly_wmma_kernel<<<grid, THREADS, 0, stream>>>(
        x, p0, p1, p2, p3, p4, p5, p6, p7, p8, p9, out, nrows);
}